// IDDiversityLoss_16020228014607
// MI455X (gfx1250) — compile-verified
//
#include <hip/hip_runtime.h>

// fp32 WMMA fragment types (gfx1250, wave32):
//  A 16x4 f32  -> 2 VGPRs/lane -> float2
//  C/D 16x16   -> 8 VGPRs/lane -> float8
typedef __attribute__((ext_vector_type(2))) float v2f;
typedef __attribute__((ext_vector_type(8))) float v8f;

#define N_ROWS 8192
#define DIMS   512
#define EPS_F  1e-8f
#define MT     64                 // macro-tile edge (4x4 WMMA tiles)
#define NB     (N_ROWS / MT)      // 128 macro-blocks per edge
#define NTRI   (NB * (NB + 1) / 2)

// ---------------------------------------------------------------- norms ----
__global__ __launch_bounds__(256) void idiv_norms_kernel(
    const float* __restrict__ A, float* __restrict__ norms) {
  const int wave = threadIdx.x >> 5;
  const int lane = threadIdx.x & 31;
  const int row  = blockIdx.x * 8 + wave;
  if (row >= N_ROWS) return;
  const float* p = A + (size_t)row * DIMS + lane;
  float s = 0.0f;
#pragma unroll
  for (int t = 0; t < DIMS / 32; ++t) {
    float v = p[t * 32];
    s = fmaf(v, v, s);
  }
#pragma unroll
  for (int off = 16; off > 0; off >>= 1) s += __shfl_xor(s, off, 32);
  if (lane == 0) norms[row] = sqrtf(s);
}

// ---------------------------------------------------------------- init -----
__global__ void idiv_zero_out_kernel(float* __restrict__ out) {
  if (threadIdx.x < 2) out[threadIdx.x] = 0.0f;
}

// ------------------------------------------------------------- main GEMM ---
// One wave32 per 64x64 macro-tile of the Gram matrix A * A^T.
// Only upper-triangular macro-blocks are computed; strictly-upper blocks
// are weighted x2 (relu(sim) is symmetric), diagonal blocks zero i==j.
__global__ __launch_bounds__(32) void idiv_cosloss_kernel(
    const float* __restrict__ A, const float* __restrict__ norms,
    float* __restrict__ out) {
  // Decode triangular block index -> (rb, cb), rb <= cb (scalar loop, SALU).
  int u = blockIdx.x;
  int rb = 0;
  while (u >= NB - rb) { u -= NB - rb; ++rb; }
  const int cb = rb + u;
  const int rowBase = rb * MT;
  const int colBase = cb * MT;

  const int lane = threadIdx.x;
  const int m   = lane & 15;        // row (A) / col (B) within 16-tile
  const int kof = (lane >> 4) * 2;  // K sub-offset: lanes 0-15 -> k0,k1; 16-31 -> k2,k3

  // Per-lane base pointers for the four 16-row fragments on each side.
  const float* aPtr[4];
  const float* bPtr[4];
#pragma unroll
  for (int i = 0; i < 4; ++i) {
    aPtr[i] = A + (size_t)(rowBase + i * 16 + m) * DIMS + kof;
    bPtr[i] = A + (size_t)(colBase + i * 16 + m) * DIMS + kof;  // B = A^T
  }

  const v8f vzero = {};
  v8f acc[4][4];
#pragma unroll
  for (int i = 0; i < 4; ++i)
#pragma unroll
    for (int j = 0; j < 4; ++j) acc[i][j] = vzero;

  // K loop: 128 steps of K=4; 8x global_load_b64 feeding 16 WMMAs per step.
#pragma unroll 2
  for (int k = 0; k < DIMS; k += 4) {
    v2f af[4], bf[4];
#pragma unroll
    for (int i = 0; i < 4; ++i) af[i] = *(const v2f*)(aPtr[i] + k);
#pragma unroll
    for (int j = 0; j < 4; ++j) bf[j] = *(const v2f*)(bPtr[j] + k);
#pragma unroll
    for (int i = 0; i < 4; ++i)
#pragma unroll
      for (int j = 0; j < 4; ++j)
        acc[i][j] = __builtin_amdgcn_wmma_f32_16x16x4_f32(
            /*neg_a=*/false, af[i], /*neg_b=*/false, bf[j],
            /*c_mod=*/(short)0, acc[i][j],
            /*reuse_a=*/false, /*reuse_b=*/false);
  }

  // Epilogue: D-matrix layout: VGPR v, lane l -> M = v + 8*(l>>4), N = l&15.
  const int half = lane >> 4;
  const float w = (cb > rb) ? 2.0f : 1.0f;

  float colN[4];
#pragma unroll
  for (int j = 0; j < 4; ++j) colN[j] = norms[colBase + j * 16 + m];

  float partial = 0.0f;
#pragma unroll
  for (int i = 0; i < 4; ++i) {
    float rowN[8];
#pragma unroll
    for (int v = 0; v < 8; ++v)
      rowN[v] = norms[rowBase + i * 16 + v + 8 * half];
#pragma unroll
    for (int j = 0; j < 4; ++j) {
      const int gcol = colBase + j * 16 + m;
#pragma unroll
      for (int v = 0; v < 8; ++v) {
        const int grow = rowBase + i * 16 + v + 8 * half;
        const float denom = fmaxf(rowN[v] * colN[j], EPS_F);
        float val = acc[i][j][v] / denom;   // cosine similarity
        val = fmaxf(val, 0.0f);             // relu clamp
        if (grow == gcol) val = 0.0f;       // zero diagonal
        partial += val;
      }
    }
  }
  partial *= w;

#pragma unroll
  for (int off = 16; off > 0; off >>= 1) partial += __shfl_xor(partial, off, 32);

  if (lane == 0) {
    const float scale = 1.0f / ((float)N_ROWS * (float)N_ROWS);
    const float contrib = partial * scale;
    atomicAdd(&out[0], contrib);  // total_loss (DIV_COEF == 1.0)
    atomicAdd(&out[1], contrib);  // l_id_div
  }
}

// ---------------------------------------------------------------- launch ---
extern "C" void kernel_launch(void* const* d_in, const int* in_sizes, int n_in,
                              void* d_out, int out_size, void* d_ws,
                              size_t ws_size, hipStream_t stream) {
  (void)in_sizes; (void)n_in; (void)out_size; (void)ws_size;
  const float* A = (const float*)d_in[0];
  float* out   = (float*)d_out;
  float* norms = (float*)d_ws;  // 8192 floats = 32 KB scratch

  idiv_zero_out_kernel<<<1, 32, 0, stream>>>(out);
  idiv_norms_kernel<<<N_ROWS / 8, 256, 0, stream>>>(A, norms);
  idiv_cosloss_kernel<<<NTRI, 32, 0, stream>>>(A, norms, out);
}